// SpanBasedNERModel_40613210751012
// MI455X (gfx1250) — compile-verified
//
#include <hip/hip_runtime.h>

#define SEQ_LEN    512
#define EMBED_DIM  768
#define NUM_LABELS 9
#define N_SPANS    32768
#define WIDTH_DIM  50
#define W_COLS     2354   // 3*768 + 50

typedef __attribute__((ext_vector_type(2))) float v2f;
typedef __attribute__((ext_vector_type(8))) float v8f;

// ---------------------------------------------------------------------------
// Kernel 1: repack W into Wcat[32][768] (column-contiguous over k, cols 27..31
// zero-padded) and precompute Q[w][j] = b[j] + W4[j] . width_table[w].
// ---------------------------------------------------------------------------
__global__ void ner_prep_kernel(const float* __restrict__ W,
                                const float* __restrict__ wtab,
                                const float* __restrict__ b,
                                float* __restrict__ Wcat,
                                float* __restrict__ Q) {
    int tid = blockIdx.x * blockDim.x + threadIdx.x;
    const int total = 32 * EMBED_DIM;
    if (tid < total) {
        int n = tid / EMBED_DIM;   // output column 0..31
        int k = tid % EMBED_DIM;
        float v = 0.0f;
        if (n < 9)        v = W[n * W_COLS + k];                    // start block
        else if (n < 18)  v = W[(n - 9) * W_COLS + 768 + k];        // end block
        else if (n < 27)  v = W[(n - 18) * W_COLS + 1536 + k];      // avg block
        Wcat[n * EMBED_DIM + k] = v;                                // 27..31 = 0
    } else {
        int t = tid - total;
        if (t < 17 * NUM_LABELS) {
            int w = t / NUM_LABELS, j = t % NUM_LABELS;
            float acc = b[j];
            const float* wr = W + j * W_COLS + 2304;
            const float* we = wtab + w * WIDTH_DIM;
            #pragma unroll 10
            for (int c = 0; c < WIDTH_DIM; ++c) acc += wr[c] * we[c];
            Q[w * 16 + j] = acc;
        }
    }
}

// ---------------------------------------------------------------------------
// Kernel 2: P[512][32] = seq[512][768] @ Wcat^T via V_WMMA_F32_16X16X4_F32.
// One block = 16 rows of seq; wave 0 -> N-tile 0 (cols 0..15), wave 1 -> cols
// 16..31. fp32 WMMA keeps exact precision vs the fp32 reference.
//
// A 16x4 f32 layout: lanes 0-15 hold M=lane (K=0 in v0, K=1 in v1);
//                    lanes 16-31 hold M=lane-16 (K=2,3).
// B 4x16 f32 layout (mirror): lanes 0-15 hold N=lane (K=0,1);
//                             lanes 16-31 hold N=lane-16 (K=2,3).
// C/D 16x16 f32: lane<16 -> (M=r, N=lane); lane>=16 -> (M=8+r, N=lane-16).
// ---------------------------------------------------------------------------
__global__ void ner_proj_wmma_kernel(const float* __restrict__ seq,
                                     const float* __restrict__ Wcat,
                                     float* __restrict__ P) {
    const int lane = threadIdx.x & 31;
    const int nt   = threadIdx.x >> 5;       // N tile: 0 or 1
    const int m0   = blockIdx.x * 16;
    const int half = lane & 15;
    const int koff = (lane < 16) ? 0 : 2;

    const float* arow = seq  + (m0 + half) * EMBED_DIM + koff;           // row-major seq
    const float* brow = Wcat + (nt * 16 + half) * EMBED_DIM + koff;      // col-contiguous

    v8f acc = {};
    #pragma unroll 8
    for (int k = 0; k < EMBED_DIM; k += 4) {
        v2f a  = *(const v2f*)(arow + k);
        v2f bb = *(const v2f*)(brow + k);
        acc = __builtin_amdgcn_wmma_f32_16x16x4_f32(
            /*neg_a=*/false, a, /*neg_b=*/false, bb,
            /*c_mod=*/(short)0, acc, /*reuse_a=*/false, /*reuse_b=*/false);
    }

    const int mBase = m0 + ((lane < 16) ? 0 : 8);
    const int nOut  = nt * 16 + half;
    #pragma unroll
    for (int r = 0; r < 8; ++r)
        P[(mBase + r) * 32 + nOut] = acc[r];
}

// ---------------------------------------------------------------------------
// Kernel 3: exclusive prefix of the 9 "avg" columns of P.
// Hillis-Steele inclusive scan in LDS (9 log-steps, 18 KB LDS) instead of a
// 512-deep serial global-latency chain. C3[0][j]=0, C3[i+1][j]=incl[i][j].
// ---------------------------------------------------------------------------
__global__ void ner_prefix_kernel(const float* __restrict__ P,
                                  float* __restrict__ C3) {
    __shared__ float sc[NUM_LABELS][SEQ_LEN];
    const int i = threadIdx.x;   // 0..511

    #pragma unroll
    for (int j = 0; j < NUM_LABELS; ++j)
        sc[j][i] = P[i * 32 + 18 + j];
    __syncthreads();

    #pragma unroll
    for (int off = 1; off < SEQ_LEN; off <<= 1) {
        float t[NUM_LABELS];
        const bool act = (i >= off);
        #pragma unroll
        for (int j = 0; j < NUM_LABELS; ++j)
            t[j] = act ? sc[j][i - off] : 0.0f;
        __syncthreads();
        if (act) {
            #pragma unroll
            for (int j = 0; j < NUM_LABELS; ++j)
                sc[j][i] += t[j];
        }
        __syncthreads();
    }

    if (i == 0) {
        #pragma unroll
        for (int j = 0; j < NUM_LABELS; ++j) C3[j] = 0.0f;
    }
    #pragma unroll
    for (int j = 0; j < NUM_LABELS; ++j)
        C3[(i + 1) * 16 + j] = sc[j][i];
}

// ---------------------------------------------------------------------------
// Kernel 4: per-span combine. Tables (P:64KB, C3:33KB, Q:1KB) are L2-resident;
// the 1.18 MB output write dominates, so stage results through LDS and emit
// unit-stride, fully coalesced stores (each block owns a contiguous 9216 B).
// ---------------------------------------------------------------------------
__global__ void ner_span_kernel(const int* __restrict__ sidx,
                                const int* __restrict__ eidx,
                                const int* __restrict__ widths,
                                const float* __restrict__ P,
                                const float* __restrict__ C3,
                                const float* __restrict__ Q,
                                float* __restrict__ out) {
    __shared__ float stage[256 * NUM_LABELS];   // 9216 B
    const int t = threadIdx.x;
    const int n = blockIdx.x * 256 + t;         // N_SPANS % 256 == 0: no guard

    const int s = sidx[n], e = eidx[n], w = widths[n];
    const float inv = 1.0f / (float)w;
    const float* Ps = P  + s * 32;       // start-projection, cols 0..8
    const float* Pe = P  + e * 32 + 9;   // end-projection,   cols 9..17
    const float* Cs = C3 + s * 16;
    const float* Ce = C3 + e * 16;
    const float* Qw = Q  + w * 16;

    #pragma unroll
    for (int j = 0; j < NUM_LABELS; ++j)         // stride-9 LDS writes: 9 ⟂ 64 banks
        stage[t * NUM_LABELS + j] = Ps[j] + Pe[j] + (Ce[j] - Cs[j]) * inv + Qw[j];
    __syncthreads();

    float* ob = out + (size_t)blockIdx.x * 256 * NUM_LABELS;
    #pragma unroll
    for (int k = 0; k < NUM_LABELS; ++k)         // unit-stride coalesced stores
        ob[k * 256 + t] = stage[k * 256 + t];
}

// ---------------------------------------------------------------------------
extern "C" void kernel_launch(void* const* d_in, const int* in_sizes, int n_in,
                              void* d_out, int out_size, void* d_ws, size_t ws_size,
                              hipStream_t stream) {
    const float* seq  = (const float*)d_in[0];   // [1,512,768]
    const int*   sidx = (const int*)d_in[1];     // [32768]
    const int*   eidx = (const int*)d_in[2];     // [32768]
    const int*   wid  = (const int*)d_in[3];     // [32768]
    const float* wtab = (const float*)d_in[4];   // [17,50]
    const float* W    = (const float*)d_in[5];   // [9,2354]
    const float* b    = (const float*)d_in[6];   // [9]
    float* out = (float*)d_out;                  // [32768,9]

    float* ws   = (float*)d_ws;
    float* Wcat = ws;                            // 32*768   = 24576 floats
    float* P    = Wcat + 32 * EMBED_DIM;         // 512*32   = 16384 floats
    float* C3   = P + SEQ_LEN * 32;              // 513*16   =  8208 floats
    float* Q    = C3 + (SEQ_LEN + 1) * 16;       // 17*16    =   272 floats

    // prep: 24576 repack threads + 153 Q threads -> 97 blocks of 256
    ner_prep_kernel<<<97, 256, 0, stream>>>(W, wtab, b, Wcat, Q);

    // fp32 WMMA projection: 32 M-tiles, 2 waves/block (one per N-tile)
    ner_proj_wmma_kernel<<<32, 64, 0, stream>>>(seq, Wcat, P);

    // parallel LDS scan over the 9 "avg" columns
    ner_prefix_kernel<<<1, SEQ_LEN, 0, stream>>>(P, C3);

    // per-span combine with coalesced staged stores
    ner_span_kernel<<<N_SPANS / 256, 256, 0, stream>>>(
        sidx, eidx, wid, P, C3, Q, out);
}